// TritonUltimateAugment_47399259079062
// MI455X (gfx1250) — compile-verified
//
#include <hip/hip_runtime.h>
#include <cstdint>

// Fused crop(8,12 -> 192x192) + hflip + brightness/contrast + saturation +
// clip + normalize for input (256,3,224,224) f32 -> output (256,3,192,192) f32.
//
// Memory-bound: ~226 MB total traffic -> ~9.7us floor at 23.3 TB/s.
// CDNA5 path: double-buffered global_load_async_to_lds_b128 pipeline with
// partial s_wait_asynccnt waits, ds_load_b128 readback, NT 128b stores.
// Each thread handles ITER=4 consecutive rows to amortize index math.

typedef float v4f __attribute__((ext_vector_type(4)));

#define BLOCK 256
#define ITER 4

#define ASYNC_LD3(ldsoff, ptr)                                              \
  do {                                                                      \
    asm volatile("global_load_async_to_lds_b128 %0, %1, off"                \
                 :: "v"(ldsoff),        "v"(ptr)                 : "memory");\
    asm volatile("global_load_async_to_lds_b128 %0, %1, off"                \
                 :: "v"((ldsoff) + 16u), "v"((ptr) + PLANE_IN)   : "memory");\
    asm volatile("global_load_async_to_lds_b128 %0, %1, off"                \
                 :: "v"((ldsoff) + 32u), "v"((ptr) + 2 * PLANE_IN): "memory");\
  } while (0)

__device__ __forceinline__ v4f rev4(v4f v) {
  v4f o; o.x = v.w; o.y = v.z; o.z = v.y; o.w = v.x; return o;
}

__device__ __forceinline__ v4f clamp01(v4f v) {
  v4f o;
  o.x = fminf(fmaxf(v.x, 0.0f), 1.0f);
  o.y = fminf(fmaxf(v.y, 0.0f), 1.0f);
  o.z = fminf(fmaxf(v.z, 0.0f), 1.0f);
  o.w = fminf(fmaxf(v.w, 0.0f), 1.0f);
  return o;
}

__global__ __launch_bounds__(BLOCK) void augment_kernel(
    const float* __restrict__ in, float* __restrict__ out) {
  constexpr int W4        = 48;          // 192 / 4 output float4s per row
  constexpr int HB        = 192 / ITER;  // 48 row-groups
  constexpr int WIN       = 224;
  constexpr int PLANE_IN  = 224 * 224;
  constexpr int PLANE_OUT = 192 * 192;

  __shared__ v4f tile[2][BLOCK * 3];

  const unsigned g  = blockIdx.x * (unsigned)BLOCK + threadIdx.x;
  const int tw = (int)(g % W4);
  const int hb = (int)((g / W4) % HB);
  const int n  = (int)(g / (W4 * HB));
  const int h0 = hb * ITER;

  // out cols [4tw..4tw+3] <- in cols [203-4tw .. 200-4tw]  (LEFT=12 folded in)
  const float* p = in  + (n * 3) * PLANE_IN  + (8 + h0) * WIN + (200 - 4 * tw);
  float*       q = out + (n * 3) * PLANE_OUT + h0 * 192 + 4 * tw;

  // Raw LDS byte offsets of this thread's staging slots (low 32 bits of the
  // flat address of a __shared__ object == DS address).
  const unsigned ldsA = (unsigned)(uintptr_t)(&tile[0][threadIdx.x * 3]);
  const unsigned ldsB = (unsigned)(uintptr_t)(&tile[1][threadIdx.x * 3]);

  // Prologue: issue row 0 loads into buffer A.
  ASYNC_LD3(ldsA, p);

#pragma unroll
  for (int i = 0; i < ITER; ++i) {
    if (i + 1 < ITER) {
      // Issue next row's loads into the other buffer, then wait for the
      // current row only (<=3 outstanding => current 3 have completed).
      const unsigned nxt = ((i + 1) & 1) ? ldsB : ldsA;
      ASYNC_LD3(nxt, p + WIN);
      asm volatile("s_wait_asynccnt 3" ::: "memory");
    } else {
      asm volatile("s_wait_asynccnt 0" ::: "memory");
    }

    const v4f* cur = &tile[i & 1][threadIdx.x * 3];
    v4f xr = rev4(cur[0]);
    v4f xg = rev4(cur[1]);
    v4f xb = rev4(cur[2]);

    // Brightness*contrast folded: (x*1.1 - 0.5)*0.9 + 0.5 = 0.99x + 0.05
    constexpr float A  = 1.1f * 0.9f;          // 0.99
    constexpr float Bc = 0.5f - 0.5f * 0.9f;   // 0.05
    xr = xr * A + Bc;
    xg = xg * A + Bc;
    xb = xb * A + Bc;

    // Saturation mix about luma: y = 1.2*x - 0.2*gray, then clip [0,1]
    v4f gray = 0.299f * xr + 0.587f * xg + 0.114f * xb;
    constexpr float S = 1.2f;
    v4f yr = clamp01(S * xr + (1.0f - S) * gray);
    v4f yg = clamp01(S * xg + (1.0f - S) * gray);
    v4f yb = clamp01(S * xb + (1.0f - S) * gray);

    // Normalize: (y - mean)/std = y*(1/std) - mean/std  (per channel)
    constexpr float s0 = 1.0f / 0.229f, b0 = 0.485f / 0.229f;
    constexpr float s1 = 1.0f / 0.224f, b1 = 0.456f / 0.224f;
    constexpr float s2 = 1.0f / 0.225f, b2 = 0.406f / 0.225f;
    v4f o0 = yr * s0 - b0;
    v4f o1 = yg * s1 - b1;
    v4f o2 = yb * s2 - b2;

    // Non-temporal 128-bit stores: output is write-once, keep it out of L2.
    __builtin_nontemporal_store(o0, (v4f*)(q));
    __builtin_nontemporal_store(o1, (v4f*)(q + PLANE_OUT));
    __builtin_nontemporal_store(o2, (v4f*)(q + 2 * PLANE_OUT));

    p += WIN;   // next input row
    q += 192;   // next output row
  }
}

extern "C" void kernel_launch(void* const* d_in, const int* in_sizes, int n_in,
                              void* d_out, int out_size, void* d_ws, size_t ws_size,
                              hipStream_t stream) {
  (void)in_sizes; (void)n_in; (void)out_size; (void)d_ws; (void)ws_size;
  const float* in = (const float*)d_in[0];
  float* out = (float*)d_out;

  // 256 batches * 48 row-groups * 48 col-groups = 589,824 threads
  // -> 2304 blocks of 256 (exact, no tail, EXEC always full)
  const unsigned total  = 256u * 48u * 48u;
  const unsigned blocks = total / BLOCK;
  augment_kernel<<<blocks, BLOCK, 0, stream>>>(in, out);
}